// GraphAttentionLayer_3539053052652
// MI455X (gfx1250) — compile-verified
//
#include <hip/hip_runtime.h>

typedef __attribute__((ext_vector_type(2))) float v2f;
typedef __attribute__((ext_vector_type(8))) float v8f;

constexpr int BS = 8;
constexpr int Nn = 2048;
constexpr int Fd = 256;
constexpr float NEG_BIG = -9000000000000000.0f;
constexpr float SLOPE   = 0.2f;

__device__ __forceinline__ float leaky(float t) { return t > 0.0f ? t : SLOPE * t; }

// Byte offset of a generic pointer within the wave's LDS space (addrspacecast
// to AS3, then truncate to the 32-bit LDS byte address the ISA expects).
__device__ __forceinline__ unsigned lds_off(void* p) {
    return (unsigned)(unsigned long long)(__attribute__((address_space(3))) void*)p;
}

// Direct global -> LDS async copy of one 16x256-f32 tile (16 KB) by 256
// threads: each thread issues 4 GLOBAL_LOAD_ASYNC_TO_LDS_B128 (tracked by
// ASYNCcnt, drained with s_wait_asynccnt). Data never passes through VGPRs.
__device__ __forceinline__ void stage_async(const float* gsrc, float* ldst, int tid)
{
    unsigned           l = lds_off(ldst) + (unsigned)tid * 16u;
    unsigned long long g = (unsigned long long)(const void*)gsrc + (unsigned long long)tid * 16ull;
#pragma unroll
    for (int r = 0; r < 4; ++r) {
        asm volatile("global_load_async_to_lds_b128 %0, %1, off offset:%2"
                     :: "v"(l), "v"(g), "i"(r * 4096) : "memory");
    }
}

// ---------------------------------------------------------------------------
// Kernel 1: h = x @ W   (M = BS*Nn = 16384, K = N = 256), f32 WMMA 16x16x4.
// Wave owns a 16x64 tile (4 C accumulators). A-frag layout per ISA:
//   lanes 0-15: row M=lane, v0=K0, v1=K1 ; lanes 16-31: row M=lane-16, v0=K2, v1=K3
// C/D layout: VGPR r, lanes 0-15 -> M=r ; lanes 16-31 -> M=r+8, N=lane%16.
// ---------------------------------------------------------------------------
__global__ __launch_bounds__(128)
void k_hgemm(const float* __restrict__ x, const float* __restrict__ W,
             float* __restrict__ h)
{
    const int lane = threadIdx.x & 31;
    const int wave = threadIdx.x >> 5;
    const int half = lane >> 4;
    const int l15  = lane & 15;
    const int row0 = blockIdx.x * 16;     // global row tile, [0, BS*Nn)
    const int col0 = wave * 64;           // 4 waves cover 256 cols
    const int arow = row0 + l15;
    const float* xr = x + (size_t)arow * Fd;

    v8f acc0 = {}, acc1 = {}, acc2 = {}, acc3 = {};

    for (int k0 = 0; k0 < Fd; k0 += 4) {
        const int ka = k0 + 2 * half;
        v2f a;
        a.x = xr[ka];
        a.y = xr[ka + 1];
        const float* wk = W + (size_t)ka * Fd;   // rows ka, ka+1 of W
#define BFRAG(t, A)                                                            \
        { v2f b; const int n = col0 + 16 * (t) + l15;                          \
          b.x = wk[n]; b.y = wk[Fd + n];                                       \
          A = __builtin_amdgcn_wmma_f32_16x16x4_f32(false, a, false, b,        \
                                                    (short)0, A, false, false); }
        BFRAG(0, acc0) BFRAG(1, acc1) BFRAG(2, acc2) BFRAG(3, acc3)
#undef BFRAG
    }

#pragma unroll
    for (int r = 0; r < 8; ++r) {
        const int m = r + 8 * half;
        float* hr = h + (size_t)(row0 + m) * Fd + col0 + l15;
        hr[0]  = acc0[r];
        hr[16] = acc1[r];
        hr[32] = acc2[r];
        hr[48] = acc3[r];
    }
}

// ---------------------------------------------------------------------------
// Kernel 2: s1[row] = h[row,:]·a1 ; s2[row] = h[row,:]·a2.  One wave per row.
// ---------------------------------------------------------------------------
__global__ __launch_bounds__(128)
void k_rowdot(const float* __restrict__ h, const float* __restrict__ a,
              float* __restrict__ s1, float* __restrict__ s2)
{
    const int lane = threadIdx.x & 31;
    const int row  = blockIdx.x * 4 + (threadIdx.x >> 5);   // [0, BS*Nn)
    const float* hr = h + (size_t)row * Fd;

    float d1 = 0.0f, d2 = 0.0f;
    for (int c = lane; c < Fd; c += 32) {
        const float hv = hr[c];
        d1 = fmaf(hv, a[c], d1);
        d2 = fmaf(hv, a[Fd + c], d2);
    }
#pragma unroll
    for (int off = 16; off > 0; off >>= 1) {
        d1 += __shfl_xor(d1, off, 32);
        d2 += __shfl_xor(d2, off, 32);
    }
    if (lane == 0) { s1[row] = d1; s2[row] = d2; }
}

// ---------------------------------------------------------------------------
// Kernel 3: row softmax stats. One wave per (b,i) row; online (max,sum) with
// shfl-xor pair merge. Fully masked rows degrade to uniform weights naturally.
// ---------------------------------------------------------------------------
__global__ __launch_bounds__(128)
void k_rowstats(const int* __restrict__ adj, const float* __restrict__ s1,
                const float* __restrict__ s2,
                float* __restrict__ rmax, float* __restrict__ rsum)
{
    const int lane = threadIdx.x & 31;
    const int row  = blockIdx.x * 4 + (threadIdx.x >> 5);   // [0, BS*Nn)
    const int b = row / Nn;
    const int i = row % Nn;
    const float s1i = s1[row];
    const int*   ar  = adj + (size_t)i * Nn;
    const float* s2b = s2 + (size_t)b * Nn;

    float m = NEG_BIG, s = 0.0f;
    for (int j = lane; j < Nn; j += 32) {
        const float val = (ar[j] > 0) ? leaky(s1i + s2b[j]) : NEG_BIG;
        if (val > m) { s = s * __expf(m - val) + 1.0f; m = val; }
        else         { s += __expf(val - m); }
    }
#pragma unroll
    for (int off = 16; off > 0; off >>= 1) {
        const float om = __shfl_xor(m, off, 32);
        const float os = __shfl_xor(s, off, 32);
        const float nm = fmaxf(m, om);
        s = s * __expf(m - nm) + os * __expf(om - nm);
        m = nm;
    }
    if (lane == 0) { rmax[row] = m; rsum[row] = s; }
}

// ---------------------------------------------------------------------------
// Kernel 4: out = elu(P @ h), P built on the fly from (adj, s1, s2, stats).
// Block = 256 threads = 8 waves = 2 row-tiles x 4 col-chunks -> 32 rows x 256
// cols per block. h j-tiles are double-buffered in LDS via
// GLOBAL_LOAD_ASYNC_TO_LDS_B128: tile jt+1 streams in (ASYNCcnt) while all 8
// waves run WMMAs + softmax VALU on tile jt. 134 MB attention matrix is never
// materialized (flash-attention style, two-pass softmax).
// ---------------------------------------------------------------------------
__global__ __launch_bounds__(256)
void k_attn(const float* __restrict__ h, const int* __restrict__ adj,
            const float* __restrict__ s1, const float* __restrict__ s2,
            const float* __restrict__ rmax, const float* __restrict__ rsum,
            float* __restrict__ out)
{
    __shared__ float lh[2][16 * Fd];              // 2 x 16 KB double buffer

    const int tid  = threadIdx.x;
    const int lane = tid & 31;
    const int wave = tid >> 5;
    const int half = lane >> 4;
    const int l15  = lane & 15;

    const int b   = blockIdx.x >> 6;              // 64 blocks per batch
    const int i0b = (blockIdx.x & 63) * 32;       // row offset within batch
    const int i0  = i0b + 16 * (wave >> 2);       // wave's 16-row tile
    const int c0  = 64 * (wave & 3);              // wave's 64-col chunk

    const int irow = i0 + l15;                    // [0, Nn)
    const size_t rb = (size_t)b * Nn + irow;
    const float s1i = s1[rb];
    const float mi  = rmax[rb];
    const float ri  = 1.0f / rsum[rb];
    const int*   ar  = adj + (size_t)irow * Nn;
    const float* s2b = s2 + (size_t)b * Nn;
    const float* hb  = h + (size_t)b * Nn * Fd;

    v8f acc0 = {}, acc1 = {}, acc2 = {}, acc3 = {};

    constexpr int NT = Nn / 16;                   // 128 j-tiles
    stage_async(hb, &lh[0][0], tid);              // prologue: tile 0 -> buf 0

    for (int jt = 0; jt < NT; ++jt) {
        const int cur = jt & 1;
        asm volatile("s_wait_asynccnt 0x0" ::: "memory");  // drain this wave's stage
        __syncthreads();                                    // all waves' tile visible
        if (jt + 1 < NT)                                    // prefetch next tile
            stage_async(hb + (size_t)(jt + 1) * 16 * Fd, &lh[cur ^ 1][0], tid);

        const int j0 = jt * 16;
        const float* lb = &lh[cur][0];
#pragma unroll
        for (int kc = 0; kc < 4; ++kc) {
            const int ja  = j0 + 4 * kc + 2 * half;    // this lane's two j's
            const int a0  = ar[ja];
            const int a1v = ar[ja + 1];
            const float v0 = (a0  > 0) ? leaky(s1i + s2b[ja])     : NEG_BIG;
            const float v1 = (a1v > 0) ? leaky(s1i + s2b[ja + 1]) : NEG_BIG;
            v2f p;
            p.x = __expf(v0 - mi) * ri;                // underflows to 0 when masked
            p.y = __expf(v1 - mi) * ri;

            const int kb = 4 * kc + 2 * half;          // j-local rows of B frag
            const float* l0 = lb + kb * Fd;
#define BACC(t, A)                                                             \
            { v2f bb; const int n = c0 + 16 * (t) + l15;                       \
              bb.x = l0[n]; bb.y = l0[Fd + n];                                 \
              A = __builtin_amdgcn_wmma_f32_16x16x4_f32(false, p, false, bb,   \
                                                        (short)0, A, false, false); }
            BACC(0, acc0) BACC(1, acc1) BACC(2, acc2) BACC(3, acc3)
#undef BACC
        }
        // No trailing barrier: buffer `cur` is only overwritten by the stage
        // issued in iteration jt+1, which happens after that iteration's
        // s_wait_asynccnt + __syncthreads (i.e. after every wave finished here).
    }

    // ELU epilogue + store
#pragma unroll
    for (int r = 0; r < 8; ++r) {
        const int m = r + 8 * half;
        float* orow = out + ((size_t)b * Nn + i0 + m) * Fd + c0 + l15;
        const float e0 = acc0[r], e1 = acc1[r], e2 = acc2[r], e3 = acc3[r];
        orow[0]  = e0 > 0.0f ? e0 : __expf(e0) - 1.0f;
        orow[16] = e1 > 0.0f ? e1 : __expf(e1) - 1.0f;
        orow[32] = e2 > 0.0f ? e2 : __expf(e2) - 1.0f;
        orow[48] = e3 > 0.0f ? e3 : __expf(e3) - 1.0f;
    }
}

// ---------------------------------------------------------------------------
extern "C" void kernel_launch(void* const* d_in, const int* in_sizes, int n_in,
                              void* d_out, int out_size, void* d_ws, size_t ws_size,
                              hipStream_t stream)
{
    const float* x   = (const float*)d_in[0];   // (8, 2048, 256) f32
    const int*   adj = (const int*)  d_in[1];   // (2048, 2048) i32
    const float* W   = (const float*)d_in[2];   // (256, 256) f32
    const float* a   = (const float*)d_in[3];   // (512, 1) f32
    float* out = (float*)d_out;                 // (8, 2048, 256) f32

    // workspace layout: h (16 MB) + 4 row-stat arrays (64 KB each)
    float* h    = (float*)d_ws;
    float* s1   = h    + (size_t)BS * Nn * Fd;
    float* s2   = s1   + (size_t)BS * Nn;
    float* rmax = s2   + (size_t)BS * Nn;
    float* rsum = rmax + (size_t)BS * Nn;

    k_hgemm   <<<(BS * Nn) / 16, 128, 0, stream>>>(x, W, h);
    k_rowdot  <<<(BS * Nn) / 4,  128, 0, stream>>>(h, a, s1, s2);
    k_rowstats<<<(BS * Nn) / 4,  128, 0, stream>>>(adj, s1, s2, rmax, rsum);
    k_attn    <<<BS * (Nn / 32), 256, 0, stream>>>(h, adj, s1, s2, rmax, rsum, out);
}